// FullAttention_10350871183815
// MI455X (gfx1250) — compile-verified
//
#include <hip/hip_runtime.h>

// MI455X / gfx1250, wave32. Full attention with materialized attention matrix.
// B=1,T=4,L=S=2048,H=8,E=D=64. f32 WMMA 16x16x4 for both GEMMs.

typedef float v2f __attribute__((ext_vector_type(2)));
typedef float v8f __attribute__((ext_vector_type(8)));

#define BT_  4
#define L_   2048
#define S_   2048
#define H_   8
#define E_   64
#define D_   64

__global__ __launch_bounds__(32)
void full_attn_f32_wmma(const float* __restrict__ Q,
                        const float* __restrict__ K,
                        const float* __restrict__ V,
                        const unsigned char* __restrict__ mask, // bool [L,S]
                        float* __restrict__ outV,               // [BT,L,H,D]
                        float* __restrict__ outA)               // [BT,H,L,S]
{
    const int lane = threadIdx.x;      // 0..31
    const int col  = lane & 15;        // N index / M index for A-operand
    const int g    = lane >> 4;        // lane-group (0,1)
    const int l0   = blockIdx.x * 16;  // L tile
    const int h    = blockIdx.y;
    const int bt   = blockIdx.z;

    // ---- Preload Q A-operand fragments: A[m][k], m=col, k = j + 2g, 16 steps of K=4 over E=64
    v2f qa[16];
    {
        const float* qp = Q + (((size_t)bt * L_ + (l0 + col)) * H_ + h) * E_ + 2 * g;
        #pragma unroll
        for (int i = 0; i < 16; ++i)
            qa[i] = *(const v2f*)(qp + 4 * i);
    }

    const float* kbase = K + (((size_t)bt * S_) * H_ + h) * E_; // + s*H_*E_ + e
    const float* vbase = V + (((size_t)bt * S_) * H_ + h) * D_; // + s*H_*D_ + d

    __shared__ float pl[16][17]; // padded transpose staging (bank-conflict free)

    float rsum[8];
    #pragma unroll
    for (int r = 0; r < 8; ++r) rsum[r] = 0.0f;

    // =================== Pass 1: QK^T -> exp -> per-lane row sums ===================
    for (int sc = 0; sc < S_ / 16; ++sc) {
        const int s0 = sc * 16;
        v8f acc = {0.f,0.f,0.f,0.f,0.f,0.f,0.f,0.f};
        const float* kp = kbase + (size_t)(s0 + col) * (H_ * E_) + 2 * g;
        #pragma unroll
        for (int i = 0; i < 16; ++i) {
            v2f kb = *(const v2f*)(kp + 4 * i);
            acc = __builtin_amdgcn_wmma_f32_16x16x4_f32(
                false, qa[i], false, kb, (short)0, acc, false, false);
        }
        #pragma unroll
        for (int r = 0; r < 8; ++r) {
            const int lrow = l0 + r + 8 * g;
            const float f = mask[(size_t)lrow * S_ + s0 + col] ? 0.1f : 0.9f;
            rsum[r] += __expf(acc[r] * (0.125f * f));
        }
    }

    // Cross-lane reduce within each 16-lane half (rows r and r+8 stay separate)
    #pragma unroll
    for (int off = 1; off < 16; off <<= 1) {
        #pragma unroll
        for (int r = 0; r < 8; ++r)
            rsum[r] += __shfl_xor(rsum[r], off, 32);
    }
    float rinv[8];
    #pragma unroll
    for (int r = 0; r < 8; ++r) rinv[r] = 1.0f / rsum[r];

    // =================== Pass 2: recompute, write attention, A*V ===================
    v8f vacc[4];
    #pragma unroll
    for (int nt = 0; nt < 4; ++nt)
        vacc[nt] = (v8f){0.f,0.f,0.f,0.f,0.f,0.f,0.f,0.f};

    float* aout = outA + (((size_t)bt * H_ + h) * L_ + l0) * S_;

    for (int sc = 0; sc < S_ / 16; ++sc) {
        const int s0 = sc * 16;
        v8f acc = {0.f,0.f,0.f,0.f,0.f,0.f,0.f,0.f};
        const float* kp = kbase + (size_t)(s0 + col) * (H_ * E_) + 2 * g;
        #pragma unroll
        for (int i = 0; i < 16; ++i) {
            v2f kb = *(const v2f*)(kp + 4 * i);
            acc = __builtin_amdgcn_wmma_f32_16x16x4_f32(
                false, qa[i], false, kb, (short)0, acc, false, false);
        }

        // normalized attention probs; stream to global (NT) and stage in LDS
        #pragma unroll
        for (int r = 0; r < 8; ++r) {
            const int rr = r + 8 * g;
            const float f = mask[(size_t)(l0 + rr) * S_ + s0 + col] ? 0.1f : 0.9f;
            const float p = __expf(acc[r] * (0.125f * f)) * rinv[r];
            __builtin_nontemporal_store(p, aout + (size_t)rr * S_ + s0 + col);
            pl[rr][col] = p;
        }
        __syncthreads();

        // A*V: A = p-tile [16(M=l) x 16(K=s)], B = V chunk [16(K=s) x 16(N=d)] x 4 d-tiles
        const float* vp = vbase + (size_t)s0 * (H_ * D_);
        #pragma unroll
        for (int ks = 0; ks < 4; ++ks) {
            v2f pa;
            pa.x = pl[col][ks * 4 + 2 * g];
            pa.y = pl[col][ks * 4 + 2 * g + 1];
            #pragma unroll
            for (int nt = 0; nt < 4; ++nt) {
                v2f vb;
                vb.x = vp[(size_t)(ks * 4 + 2 * g)     * (H_ * D_) + nt * 16 + col];
                vb.y = vp[(size_t)(ks * 4 + 2 * g + 1) * (H_ * D_) + nt * 16 + col];
                vacc[nt] = __builtin_amdgcn_wmma_f32_16x16x4_f32(
                    false, pa, false, vb, (short)0, vacc[nt], false, false);
            }
        }
        __syncthreads();
    }

    // Store V: out[bt][l][h][d]
    #pragma unroll
    for (int nt = 0; nt < 4; ++nt) {
        #pragma unroll
        for (int r = 0; r < 8; ++r) {
            float* op = outV + (((size_t)bt * L_ + l0 + r + 8 * g) * H_ + h) * D_
                      + nt * 16 + col;
            __builtin_nontemporal_store(vacc[nt][r], op);
        }
    }
}

extern "C" void kernel_launch(void* const* d_in, const int* in_sizes, int n_in,
                              void* d_out, int out_size, void* d_ws, size_t ws_size,
                              hipStream_t stream) {
    (void)in_sizes; (void)n_in; (void)d_ws; (void)ws_size; (void)out_size;
    const float* Q = (const float*)d_in[0];            // [1,4,2048,8,64]
    const float* K = (const float*)d_in[1];            // [1,4,2048,8,64]
    const float* V = (const float*)d_in[2];            // [1,4,2048,8,64]
    const unsigned char* mask = (const unsigned char*)d_in[3]; // bool [2048,2048]
    // d_in[4] (no_tf_genes_index) is unused by the reference computation.

    float* outV = (float*)d_out;                                   // BT*L*H*D
    float* outA = outV + (size_t)BT_ * L_ * H_ * D_;               // BT*H*L*S

    dim3 grid(L_ / 16, H_, BT_);
    dim3 block(32);
    full_attn_f32_wmma<<<grid, block, 0, stream>>>(Q, K, V, mask, outV, outA);
}